// AC_FocalLoss_41729902248250
// MI455X (gfx1250) — compile-verified
//
#include <hip/hip_runtime.h>

typedef float f4  __attribute__((ext_vector_type(4)));
typedef float v2f __attribute__((ext_vector_type(2)));
typedef float v8f __attribute__((ext_vector_type(8)));

#define HW_      (1 << 20)            // 1024*1024 spatial elements per image
#define BLK      256                  // 8 wave32s per block
#define EPB      16384                // elements per block
#define F4_ITERS (EPB / (BLK * 4))    // 16 float4 iterations per thread

// Monotonic float <-> uint mapping so integer atomic min/max == float min/max.
__device__ __forceinline__ unsigned orderF(float f) {
  unsigned u = __float_as_uint(f);
  return (u & 0x80000000u) ? ~u : (u | 0x80000000u);
}
__device__ __forceinline__ float unorderF(unsigned u) {
  unsigned b = (u & 0x80000000u) ? (u & 0x7FFFFFFFu) : ~u;
  return __uint_as_float(b);
}

__global__ void k_init(unsigned* __restrict__ minU, unsigned* __restrict__ maxU,
                       unsigned* __restrict__ cnt, int nimg) {
  int t = threadIdx.x;
  if (t < nimg) minU[t] = 0xFFFFFFFFu;          // +inf in ordered space
  if (t >= 128 && t < 128 + nimg) maxU[t - 128] = 0u;  // -inf in ordered space
  if (t == 255) cnt[0] = 0u;
}

// Pass A: fused elementwise loss/weight partial sums + per-image min/max.
__global__ __launch_bounds__(BLK) void k_main(
    const float* __restrict__ in, const float* __restrict__ tg,
    unsigned* __restrict__ minU, unsigned* __restrict__ maxU,
    float* __restrict__ lossPart, float* __restrict__ wPart) {
  const int img = blockIdx.y;
  const long long base = (long long)img * HW_ + (long long)blockIdx.x * EPB;
  const f4* in4 = (const f4*)(in + base);
  const f4* tg4 = (const f4*)(tg + base);

  float sl = 0.f, sw = 0.f;
  float mn = __int_as_float(0x7F800000);   // +inf
  float mx = __int_as_float(0xFF800000);   // -inf

#pragma unroll 4
  for (int it = 0; it < F4_ITERS; ++it) {
    const int idx = it * BLK + (int)threadIdx.x;
    f4 x4 = __builtin_nontemporal_load(in4 + idx);  // streaming: NT hint, skip cache pollution
    f4 t4 = __builtin_nontemporal_load(tg4 + idx);
#pragma unroll
    for (int j = 0; j < 4; ++j) {
      const float x = x4[j], t = t4[j];
      mn = fminf(mn, x);
      mx = fmaxf(mx, x);
      const float ax  = fabsf(x);
      const float e   = __expf(-ax);
      const float l1p = log1pf(e);
      const float pls = fminf(x, 0.f) - l1p;    // log_sigmoid(x)
      const float nls = fminf(-x, 0.f) - l1p;   // log_sigmoid(-x)
      const float r   = 1.f / (1.f + e);
      const float p   = (x >= 0.f) ? r : e * r; // sigmoid(x), stable both signs
      const float q   = 1.f - p;
      const float pw  = q * q;                  // (1-p)^GAMMA, GAMMA==2
      const float nw  = p * p;                  // p^GAMMA
      sl -= t * pw * pls + (1.f - t) * nw * nls;
      sw += t * pw + (1.f - t) * nw;
    }
  }

  // wave32 butterfly reduction
  for (int m = 16; m > 0; m >>= 1) {
    sl += __shfl_xor(sl, m, 32);
    sw += __shfl_xor(sw, m, 32);
    mn = fminf(mn, __shfl_xor(mn, m, 32));
    mx = fmaxf(mx, __shfl_xor(mx, m, 32));
  }
  __shared__ float s_sl[BLK / 32], s_sw[BLK / 32], s_mn[BLK / 32], s_mx[BLK / 32];
  const int wave = threadIdx.x >> 5;
  if ((threadIdx.x & 31) == 0) {
    s_sl[wave] = sl; s_sw[wave] = sw; s_mn[wave] = mn; s_mx[wave] = mx;
  }
  __syncthreads();
  if (threadIdx.x == 0) {
    float SL = 0.f, SW = 0.f, MN = s_mn[0], MX = s_mx[0];
#pragma unroll
    for (int w = 0; w < BLK / 32; ++w) {
      SL += s_sl[w]; SW += s_sw[w];
      MN = fminf(MN, s_mn[w]); MX = fmaxf(MX, s_mx[w]);
    }
    const int pid = img * gridDim.x + blockIdx.x;
    lossPart[pid] = SL;                 // deterministic per-block slot
    wPart[pid]    = SW;
    atomicMin(&minU[img], orderF(MN));  // order-independent -> deterministic
    atomicMax(&maxU[img], orderF(MX));
  }
}

// Pass B: count of normalized > POS_THRES, given per-image min/max.
__global__ __launch_bounds__(BLK) void k_count(
    const float* __restrict__ in,
    const unsigned* __restrict__ minU, const unsigned* __restrict__ maxU,
    unsigned* __restrict__ cnt) {
  const int img = blockIdx.y;
  const float mn = unorderF(minU[img]);
  const float mx = unorderF(maxU[img]);
  // (x-mn)/(mx-mn) > 0.8  <=>  x > mn + 0.8*(mx-mn)   (mx > mn)
  const float thres = mn + 0.8f * (mx - mn);
  const long long base = (long long)img * HW_ + (long long)blockIdx.x * EPB;
  const f4* in4 = (const f4*)(in + base);

  unsigned c = 0;
#pragma unroll 4
  for (int it = 0; it < F4_ITERS; ++it) {
    f4 x4 = __builtin_nontemporal_load(in4 + it * BLK + (int)threadIdx.x);
#pragma unroll
    for (int j = 0; j < 4; ++j) c += (x4[j] > thres) ? 1u : 0u;
  }
  for (int m = 16; m > 0; m >>= 1) c += __shfl_xor(c, m, 32);
  __shared__ unsigned s_c[BLK / 32];
  if ((threadIdx.x & 31) == 0) s_c[threadIdx.x >> 5] = c;
  __syncthreads();
  if (threadIdx.x == 0) {
    unsigned C = 0;
#pragma unroll
    for (int w = 0; w < BLK / 32; ++w) C += s_c[w];
    atomicAdd(cnt, C);                  // integer: exact & deterministic
  }
}

// Final: WMMA-based reduction of the block partials (one full wave32, EXEC all-1s).
// A = ones(16x4): D = 1*B + C accumulates every element of each 64-float chunk;
// sum of all 256 entries of D equals 16x the true total, independent of B layout.
__global__ __launch_bounds__(32) void k_final(
    const float* __restrict__ lossPart, const float* __restrict__ wPart,
    const unsigned* __restrict__ cnt, int NB, long long Ntot,
    float* __restrict__ out) {
  const int lane = threadIdx.x;
  v2f ones; ones[0] = 1.f; ones[1] = 1.f;
  v8f cl = {0.f, 0.f, 0.f, 0.f, 0.f, 0.f, 0.f, 0.f};
  v8f cw = {0.f, 0.f, 0.f, 0.f, 0.f, 0.f, 0.f, 0.f};

  const int iters = NB >> 6;            // 64 partials per WMMA
  for (int k = 0; k < iters; ++k) {
    v2f bl = *(const v2f*)(lossPart + (k << 6) + lane * 2);
    v2f bw = *(const v2f*)(wPart    + (k << 6) + lane * 2);
    cl = __builtin_amdgcn_wmma_f32_16x16x4_f32(false, ones, false, bl,
                                               (short)0, cl, false, false);
    cw = __builtin_amdgcn_wmma_f32_16x16x4_f32(false, ones, false, bw,
                                               (short)0, cw, false, false);
  }

  float tl = 0.f, tw = 0.f;
#pragma unroll
  for (int j = 0; j < 8; ++j) { tl += cl[j]; tw += cw[j]; }
  for (int m = 16; m > 0; m >>= 1) {
    tl += __shfl_xor(tl, m, 32);
    tw += __shfl_xor(tw, m, 32);
  }
  if (lane == 0) {
    const double sum_loss = (double)tl * (1.0 / 16.0);
    const double sum_w    = (double)tw * (1.0 / 16.0);
    const double frac     = 1.0 - (double)cnt[0] / (double)Ntot;  // MARGIN == 0
    const double ac       = frac * sqrt(frac);                    // ^1.5 (frac >= 0)
    out[0] = (float)((sum_loss / sum_w) * ac);
  }
}

extern "C" void kernel_launch(void* const* d_in, const int* in_sizes, int n_in,
                              void* d_out, int out_size, void* d_ws, size_t ws_size,
                              hipStream_t stream) {
  const float* in = (const float*)d_in[0];
  const float* tg = (const float*)d_in[1];
  float* out = (float*)d_out;

  const int n    = in_sizes[0];         // 64 * 1M = 67,108,864
  const int nimg = n / HW_;             // 64
  const int bpi  = HW_ / EPB;           // 64 blocks per image
  const int NB   = nimg * bpi;          // 4096 partial slots

  // Workspace layout (well under typical ws_size; ~34 KB):
  char* ws = (char*)d_ws;
  unsigned* minU    = (unsigned*)(ws + 0);
  unsigned* maxU    = (unsigned*)(ws + 512);
  unsigned* cnt     = (unsigned*)(ws + 1024);
  float*    lossP   = (float*)(ws + 2048);
  float*    wP      = (float*)(ws + 2048 + (size_t)NB * sizeof(float));

  k_init<<<1, 256, 0, stream>>>(minU, maxU, cnt, nimg);

  dim3 grid(bpi, nimg);
  k_main<<<grid, BLK, 0, stream>>>(in, tg, minU, maxU, lossP, wP);
  k_count<<<grid, BLK, 0, stream>>>(in, minU, maxU, cnt);
  k_final<<<1, 32, 0, stream>>>(lossP, wP, cnt, NB, (long long)n, out);
}